// RetrievalCosineLoss_26946624815703
// MI455X (gfx1250) — compile-verified
//
#include <hip/hip_runtime.h>

#define DIM 256         // row length (floats)
#define WPB 8           // waves per block
#define TPB 256         // threads per block (8 x wave32)

// 32-bit LDS byte address of a __shared__ object (generic -> AS3 -> int).
__device__ __forceinline__ unsigned lds_addr32(const void* p) {
  return (unsigned)(unsigned long long)(const __attribute__((address_space(3))) void*)p;
}

// Issue the async copy of one (q,d) row pair into this wave's LDS buffer.
// Per lane: 16B at +0 and 16B at +512 for each of q and d (4 instructions,
// ASYNCcnt += 4). INST_OFFSET is added to BOTH the global and LDS address,
// so offset:512 reuses the same address VGPRs for the second half-row.
__device__ __forceinline__ void issue_async_row(const float* gq, const float* gd,
                                                unsigned lq, unsigned ld) {
  asm volatile(
      "global_load_async_to_lds_b128 %0, %2, off\n\t"
      "global_load_async_to_lds_b128 %0, %2, off offset:512\n\t"
      "global_load_async_to_lds_b128 %1, %3, off\n\t"
      "global_load_async_to_lds_b128 %1, %3, off offset:512"
      :
      : "v"(lq), "v"(ld), "v"(gq), "v"(gd)
      : "memory");
}

__global__ __launch_bounds__(TPB) void RetrievalCosineLoss_26946624815703_kernel(
    const float* __restrict__ q, const float* __restrict__ d,
    float* __restrict__ out, int nrows) {
  // [wave][double-buffer][q|d][row]  -> 32 KiB per block (of 320 KiB/WGP)
  __shared__ float tile[WPB][2][2][DIM];

  const int lane = threadIdx.x & 31;
  const int wid  = threadIdx.x >> 5;
  const int step = gridDim.x * WPB;
  int r = blockIdx.x * WPB + wid;

  unsigned lq[2], ldd[2];
  lq[0]  = lds_addr32(&tile[wid][0][0][lane * 4]);
  ldd[0] = lds_addr32(&tile[wid][0][1][lane * 4]);
  lq[1]  = lds_addr32(&tile[wid][1][0][lane * 4]);
  ldd[1] = lds_addr32(&tile[wid][1][1][lane * 4]);

  // Pipeline prologue: prefetch first row into buffer 0.
  if (r < nrows)
    issue_async_row(q + (size_t)r * DIM + lane * 4,
                    d + (size_t)r * DIM + lane * 4, lq[0], ldd[0]);

  int buf = 0;
  while (r < nrows) {
    const int rn = r + step;
    const bool more = rn < nrows;

    // Prefetch next row into the other buffer, then wait for current row's
    // 4 async loads (ASYNCcnt completes in order for async loads).
    if (more) {
      issue_async_row(q + (size_t)rn * DIM + lane * 4,
                      d + (size_t)rn * DIM + lane * 4, lq[buf ^ 1], ldd[buf ^ 1]);
      asm volatile("s_wait_asynccnt 0x4" ::: "memory");
    } else {
      asm volatile("s_wait_asynccnt 0x0" ::: "memory");
    }

    const float* tq = &tile[wid][buf][0][0];
    const float* td = &tile[wid][buf][1][0];
    float4 qa = *(const float4*)(tq + lane * 8);
    float4 qb = *(const float4*)(tq + lane * 8 + 4);
    float4 da = *(const float4*)(td + lane * 8);
    float4 db = *(const float4*)(td + lane * 8 + 4);

    float qq = qa.x * qa.x + qa.y * qa.y + qa.z * qa.z + qa.w * qa.w +
               qb.x * qb.x + qb.y * qb.y + qb.z * qb.z + qb.w * qb.w;
    float dd = da.x * da.x + da.y * da.y + da.z * da.z + da.w * da.w +
               db.x * db.x + db.y * db.y + db.z * db.z + db.w * db.w;
    float qd = qa.x * da.x + qa.y * da.y + qa.z * da.z + qa.w * da.w +
               qb.x * db.x + qb.y * db.y + qb.z * db.z + qb.w * db.w;

    // wave32 butterfly reduction of the three partials
#pragma unroll
    for (int m = 16; m >= 1; m >>= 1) {
      qq += __shfl_xor(qq, m, 32);
      dd += __shfl_xor(dd, m, 32);
      qd += __shfl_xor(qd, m, 32);
    }

    if (lane == 0) out[r] = qd / (sqrtf(qq) * sqrtf(dd));

    r = rn;
    buf ^= 1;
  }
}

extern "C" void kernel_launch(void* const* d_in, const int* in_sizes, int n_in,
                              void* d_out, int out_size, void* d_ws, size_t ws_size,
                              hipStream_t stream) {
  const float* q  = (const float*)d_in[0];
  const float* dv = (const float*)d_in[1];
  float* out = (float*)d_out;
  const int nrows = in_sizes[0] / DIM;   // 262144 for the reference shapes

  int blocks = (nrows + WPB - 1) / WPB;  // 1 row per wave minimum
  if (blocks > 2048) blocks = 2048;      // 16384 waves -> 16 rows/wave, deep pipeline
  if (blocks < 1) blocks = 1;

  RetrievalCosineLoss_26946624815703_kernel<<<blocks, TPB, 0, stream>>>(q, dv, out, nrows);
}